// HyGAttEmb_25589415150167
// MI455X (gfx1250) — compile-verified
//
#include <hip/hip_runtime.h>
#include <hip/hip_bf16.h>
#include <math.h>

#define EMB 128

typedef float v2f __attribute__((ext_vector_type(2)));
typedef float v8f __attribute__((ext_vector_type(8)));

// ---------------- fill ----------------
__global__ void fill_kernel(float* __restrict__ p, float v, int n) {
    int i = blockIdx.x * blockDim.x + threadIdx.x;
    if (i < n) p[i] = v;
}

// ---------------- attention scores via V_WMMA_F32_16X16X4_F32 ----------------
// y[i] = dot(M[i,:], w)  (+ dot(M2[i,:], w2) if M2 != null), fp32 exact.
// One wave handles 16 rows. A(16x4) layout per ISA: lanes 0-15 rows, VGPR0=K0,
// VGPR1=K1; lanes 16-31 same rows, VGPR0=K2, VGPR1=K3. B(4x16) mirrors this
// with rows striped across lane halves; we broadcast w across all 16 columns,
// so D[i,j] = y[i] for every j. Extract from lane 0 (M=0..7) and lane 16 (M=8..15).
__global__ void scores_wmma_kernel(const float* __restrict__ M, const float* __restrict__ w,
                                   const float* __restrict__ M2, const float* __restrict__ w2,
                                   int n, float* __restrict__ y) {
    const int lane  = threadIdx.x & 31;
    const int wave  = threadIdx.x >> 5;
    const int tile  = blockIdx.x * (blockDim.x >> 5) + wave;  // 16 rows per wave
    const int row16 = tile * 16;
    int row = row16 + (lane & 15);
    if (row >= n) row = n - 1;               // clamp: keep EXEC all-ones for WMMA
    const int khalf = (lane >> 4) << 1;      // 0 for lanes 0-15, 2 for lanes 16-31

    v8f acc = {};
    const float* Mr = M + (size_t)row * EMB;
    #pragma unroll
    for (int c = 0; c < EMB / 4; ++c) {
        const int k = 4 * c + khalf;
        v2f a; a.x = Mr[k]; a.y = Mr[k + 1];
        v2f b; b.x = w[k];  b.y = w[k + 1];
        acc = __builtin_amdgcn_wmma_f32_16x16x4_f32(false, a, false, b, (short)0, acc,
                                                    false, false);
    }
    if (M2) {
        const float* M2r = M2 + (size_t)row * EMB;
        #pragma unroll
        for (int c = 0; c < EMB / 4; ++c) {
            const int k = 4 * c + khalf;
            v2f a; a.x = M2r[k]; a.y = M2r[k + 1];
            v2f b; b.x = w2[k];  b.y = w2[k + 1];
            acc = __builtin_amdgcn_wmma_f32_16x16x4_f32(false, a, false, b, (short)0, acc,
                                                        false, false);
        }
    }

    if ((lane & 15) == 0) {                  // lanes 0 and 16 write 8 rows each
        const int base = row16 + ((lane >> 4) << 3);
        #pragma unroll
        for (int r = 0; r < 8; ++r) {
            const int o = base + r;
            if (o < n) y[o] = acc[r];
        }
    }
}

// ---------------- per-incidence logits + segment max ----------------
__device__ __forceinline__ void atomicMaxF(float* addr, float val) {
    // sign-split trick; valid with init = -INF
    if (val >= 0.0f) atomicMax((int*)addr, __float_as_int(val));
    else             atomicMin((unsigned int*)addr, __float_as_uint(val));
}

__global__ void logits_max_kernel(const int* __restrict__ ve, const float* __restrict__ svt,
                                  const float* __restrict__ se, float* __restrict__ a,
                                  float* __restrict__ segm, int n_inc, int dst_is_e,
                                  float inv_tau) {
    int i = blockIdx.x * blockDim.x + threadIdx.x;
    if (i >= n_inc) return;
    const int v = ve[2 * i], e = ve[2 * i + 1];
    const float t = tanhf(svt[v] + se[e]) * inv_tau;
    a[i] = t;
    atomicMaxF(&segm[dst_is_e ? e : v], t);
}

// ---------------- exp + segment sum ----------------
__global__ void exp_sum_kernel(const int* __restrict__ ve, float* __restrict__ a,
                               const float* __restrict__ segm, float* __restrict__ segs,
                               int n_inc, int dst_is_e) {
    int i = blockIdx.x * blockDim.x + threadIdx.x;
    if (i >= n_inc) return;
    const int dst = dst_is_e ? ve[2 * i + 1] : ve[2 * i];
    const float ex = __expf(a[i] - segm[dst]);
    a[i] = ex;
    atomicAdd(&segs[dst], ex);
}

// ---------------- weighted gather/scatter (the bandwidth-bound core) ----------------
// one wave per incidence: 32 lanes x float4 = 128-dim row, native f32 atomics
__global__ void scatter_kernel(const int* __restrict__ ve, const float* __restrict__ a,
                               const float* __restrict__ segs, const float* __restrict__ src,
                               float* __restrict__ out, int n_inc, int dst_is_e) {
    const int lane = threadIdx.x & 31;
    const int wave = threadIdx.x >> 5;
    const int i = blockIdx.x * (blockDim.x >> 5) + wave;
    if (i >= n_inc) return;
    const int v = ve[2 * i], e = ve[2 * i + 1];
    const int dst = dst_is_e ? e : v;
    const int s   = dst_is_e ? v : e;
    const float w = a[i] / segs[dst];
    const float4 sv = ((const float4*)(src + (size_t)s * EMB))[lane];
    float* op = out + (size_t)dst * EMB + lane * 4;
    atomicAdd(op + 0, w * sv.x);
    atomicAdd(op + 1, w * sv.y);
    atomicAdd(op + 2, w * sv.z);
    atomicAdd(op + 3, w * sv.w);
}

// ---------------- relu in place + accumulate into output mean buffer ----------------
__global__ void relu_accum_kernel(float* __restrict__ buf, float* __restrict__ acc, int n4) {
    int i = blockIdx.x * blockDim.x + threadIdx.x;
    if (i >= n4) return;
    float4 x = ((float4*)buf)[i];
    x.x = fmaxf(x.x, 0.0f); x.y = fmaxf(x.y, 0.0f);
    x.z = fmaxf(x.z, 0.0f); x.w = fmaxf(x.w, 0.0f);
    ((float4*)buf)[i] = x;
    float4 o = ((float4*)acc)[i];
    o.x += x.x; o.y += x.y; o.z += x.z; o.w += x.w;
    ((float4*)acc)[i] = o;
}

__global__ void scale_kernel(float* __restrict__ p, float s, int n4) {
    int i = blockIdx.x * blockDim.x + threadIdx.x;
    if (i >= n4) return;
    float4 x = ((float4*)p)[i];
    x.x *= s; x.y *= s; x.z *= s; x.w *= s;
    ((float4*)p)[i] = x;
}

extern "C" void kernel_launch(void* const* d_in, const int* in_sizes, int n_in,
                              void* d_out, int out_size, void* d_ws, size_t ws_size,
                              hipStream_t stream) {
    const float* v_emb  = (const float*)d_in[0];
    const float* t_emb  = (const float*)d_in[1];
    const float* e_emb  = (const float*)d_in[2];
    const int*   ve     = (const int*)d_in[3];
    const float* av_v2e = (const float*)d_in[4];
    const float* at_v2e = (const float*)d_in[5];
    const float* ae_v2e = (const float*)d_in[6];
    const float* av_e2v = (const float*)d_in[7];
    const float* at_e2v = (const float*)d_in[8];
    const float* ae_e2v = (const float*)d_in[9];

    const int V  = in_sizes[0] / EMB;
    const int E  = in_sizes[2] / EMB;
    const int NI = in_sizes[3] / 2;
    const int SEG = (V > E) ? V : E;

    float* out_v = (float*)d_out;
    float* out_e = out_v + (size_t)V * EMB;

    // ---- workspace carve (all chunks 16B-aligned) ----
    char* ws = (char*)d_ws;
    float* v_buf = (float*)ws; ws += (size_t)V * EMB * sizeof(float);
    float* e_buf = (float*)ws; ws += (size_t)E * EMB * sizeof(float);
    float* svt   = (float*)ws; ws += (size_t)V * sizeof(float);
    float* se    = (float*)ws; ws += (size_t)E * sizeof(float);
    float* aw    = (float*)ws; ws += (size_t)NI * sizeof(float);
    float* segm  = (float*)ws; ws += (size_t)SEG * sizeof(float);
    float* segs  = (float*)ws; ws += (size_t)SEG * sizeof(float);

    // output accumulators start as the layer-0 stack entries
    hipMemcpyAsync(out_v, v_emb, (size_t)V * EMB * sizeof(float),
                   hipMemcpyDeviceToDevice, stream);
    hipMemcpyAsync(out_e, e_emb, (size_t)E * EMB * sizeof(float),
                   hipMemcpyDeviceToDevice, stream);

    const float* v_fea = v_emb;   // current layer v features
    const float* e_fea = e_emb;   // current layer e features

    auto B1 = [](long n) { return (int)((n + 255) / 256); };

    for (int l = 0; l < 2; ++l) {
        const float* avl = av_v2e + l * EMB; const float* atl = at_v2e + l * EMB;
        const float* ael = ae_v2e + l * EMB;
        const float* av2 = av_e2v + l * EMB; const float* at2 = at_e2v + l * EMB;
        const float* ae2 = ae_e2v + l * EMB;

        // ---- v -> e aggregation (dst = e) ----
        scores_wmma_kernel<<<(V + 127) / 128, 256, 0, stream>>>(v_fea, avl, t_emb, atl, V, svt);
        scores_wmma_kernel<<<(E + 127) / 128, 256, 0, stream>>>(e_fea, ael, nullptr, nullptr, E, se);
        fill_kernel<<<B1(E), 256, 0, stream>>>(segm, -INFINITY, E);
        fill_kernel<<<B1(E), 256, 0, stream>>>(segs, 0.0f, E);
        fill_kernel<<<B1((long)E * EMB), 256, 0, stream>>>(e_buf, 0.0f, E * EMB);
        logits_max_kernel<<<B1(NI), 256, 0, stream>>>(ve, svt, se, aw, segm, NI, 1, 1.0f);
        exp_sum_kernel<<<B1(NI), 256, 0, stream>>>(ve, aw, segm, segs, NI, 1);
        scatter_kernel<<<(NI + 7) / 8, 256, 0, stream>>>(ve, aw, segs, v_fea, e_buf, NI, 1);

        // ---- e -> v aggregation (dst = v), uses PRE-relu e_new (= e_buf) ----
        scores_wmma_kernel<<<(V + 127) / 128, 256, 0, stream>>>(v_fea, av2, t_emb, at2, V, svt);
        scores_wmma_kernel<<<(E + 127) / 128, 256, 0, stream>>>(e_buf, ae2, nullptr, nullptr, E, se);
        fill_kernel<<<B1(V), 256, 0, stream>>>(segm, -INFINITY, V);
        fill_kernel<<<B1(V), 256, 0, stream>>>(segs, 0.0f, V);
        fill_kernel<<<B1((long)V * EMB), 256, 0, stream>>>(v_buf, 0.0f, V * EMB);
        logits_max_kernel<<<B1(NI), 256, 0, stream>>>(ve, svt, se, aw, segm, NI, 0, 1.0f);
        exp_sum_kernel<<<B1(NI), 256, 0, stream>>>(ve, aw, segm, segs, NI, 0);
        scatter_kernel<<<(NI + 7) / 8, 256, 0, stream>>>(ve, aw, segs, e_buf, v_buf, NI, 0);

        // ---- relu in place (becomes next layer's features) + accumulate mean ----
        relu_accum_kernel<<<B1((long)E * EMB / 4), 256, 0, stream>>>(e_buf, out_e, E * EMB / 4);
        relu_accum_kernel<<<B1((long)V * EMB / 4), 256, 0, stream>>>(v_buf, out_v, V * EMB / 4);

        v_fea = v_buf;
        e_fea = e_buf;
    }

    // mean over 3 stack entries
    scale_kernel<<<B1(((long)V + E) * EMB / 4), 256, 0, stream>>>(out_v, 1.0f / 3.0f,
                                                                  (V + E) * EMB / 4);
}